// BrainModule_38405597561583
// MI455X (gfx1250) — compile-verified
//
#include <hip/hip_runtime.h>
#include <hip/hip_bf16.h>

typedef __attribute__((ext_vector_type(2))) float v2f;
typedef __attribute__((ext_vector_type(8))) float v8f;

#define BB    32
#define CC    273     // channels
#define CHOUT 270
#define TT    4096
#define DD    242     // actual pos dim (2 * 11 * 11)

// ---------------------------------------------------------------------------
// Kernel 1: Fourier embedding, stored TRANSPOSED: emb_t[b][d][c]
// so the scores GEMM B-operand (K=d, N=c) loads are coalesced.
// ---------------------------------------------------------------------------
__global__ __launch_bounds__(288)
void emb_kernel(const float* __restrict__ positions, float* __restrict__ emb_t) {
  int bd = blockIdx.x;
  int b = bd / DD, d = bd % DD;
  int c = threadIdx.x;
  if (c >= CC) return;
  float x = positions[((long)b * CC + c) * 2 + 0] + 0.2f;  // + MARGIN
  float y = positions[((long)b * CC + c) * 2 + 1] + 0.2f;
  int f = (d < 121) ? d : (d - 121);
  int i = f / 11, j = f - i * 11;
  const float scale = 6.28318530717958647692f / 1.4f;      // 2*pi / (1 + 2*MARGIN)
  float loc = x * (scale * (float)i) + y * (scale * (float)j);
  float v = (d < 121) ? cosf(loc) : sinf(loc);
  emb_t[((long)b * DD + d) * CC + c] = v;
}

// ---------------------------------------------------------------------------
// Batched fp32 WMMA GEMM: C[bz] = A[idx(bz)] (MxK) * B[bz] (KxN).
// All dims/strides are template constants so the K-loop strength-reduces to
// pointer increments + immediate offsets (no v_mul_u64 in the hot loop).
// Block = 256 threads = 8 waves.  Wave tile = 16(M) x 64(N): 4 accumulators
// per wave reuse one A fragment -> 4 v_wmma per K-step of 4.
// Block tile = 16(M) x 512(N).
// When N is divisible by the block tile (the dominant meg GEMM, N=4096),
// a single base pointer serves all 8 B loads via immediate offsets
// (sub-tiles 64 B apart, row stride LDB*4 <= 16 KB fits imm24).
// A tile staged in LDS, zero-padded for m>=M / k>=K, so padded WMMA lanes
// contribute exactly zero; tail/edge B loads clamp addresses.
// ---------------------------------------------------------------------------
template<int LDA, int LDB, int LDC, int M, int N, int K, bool USE_IDX>
__global__ __launch_bounds__(256)
void wmma_gemm_f32(const float* __restrict__ A, const int* __restrict__ aIdx,
                   const float* __restrict__ Bm, float* __restrict__ Cm) {
  constexpr int Kpad  = (K + 3) & ~3;
  constexpr int Kmain = K & ~3;                  // clamp-free K-steps
  constexpr int NSUB  = 4;                       // 16x16 tiles per wave
  constexpr int BLKN  = 16 * NSUB * 8;           // 512 cols per block
  constexpr bool NDIV = (N % BLKN) == 0;         // no column edge handling
  __shared__ float Alds[Kpad * 16];              // k-major: Alds[k*16 + m]

  const int bz = blockIdx.z;
  const int mt = blockIdx.y;                     // 16-row M tile
  const float* Ab = A + (long)(USE_IDX ? aIdx[bz] : bz) * ((long)M * LDA);
  const float* Bb = Bm + (long)bz * ((long)K * LDB);
  float*       Cb = Cm + (long)bz * ((long)M * LDC);

  // Stage A tile (16 rows x Kpad cols) into LDS, zero-padded.
  for (int idx = threadIdx.x; idx < Kpad * 16; idx += 256) {
    int k = idx >> 4, m = idx & 15;
    int gm = mt * 16 + m;
    Alds[idx] = (gm < M && k < K) ? Ab[(long)gm * LDA + k] : 0.0f;
  }
  __syncthreads();

  const int wave = threadIdx.x >> 5;
  const int lane = threadIdx.x & 31;
  const int nbase = blockIdx.x * BLKN + wave * (16 * NSUB);
  if (nbase >= N) return;                        // wave-uniform; no barriers after

  const int m16 = lane & 15;                     // m for A frag, n for B/C frags
  const int kh  = lane >> 4;                     // k-pair selector / m-half for C

  int ncl[NSUB];                                 // clamped load columns (edge path)
#pragma unroll
  for (int j = 0; j < NSUB; ++j) {
    int c0 = nbase + 16 * j + m16;
    ncl[j] = (c0 < N) ? c0 : (N - 1);
  }

  v8f acc[NSUB];
  const v8f vzero = {0.f, 0.f, 0.f, 0.f, 0.f, 0.f, 0.f, 0.f};
#pragma unroll
  for (int j = 0; j < NSUB; ++j) acc[j] = vzero;

  // -------- main loop: provably in-range, pure pointer increments ----------
  if constexpr (NDIV) {
    // One base pointer; 8 loads per step via immediate offsets.
    const float* Bl = Bb + (long)(2 * kh) * LDB + (nbase + m16);
    for (int k0 = 0; k0 < Kmain; k0 += 4) {
      v2f a;
      a[0] = Alds[(k0 + 2 * kh) * 16 + m16];
      a[1] = Alds[(k0 + 2 * kh + 1) * 16 + m16];
      if (k0 + 19 < K)
        __builtin_prefetch(Bl + (long)16 * LDB, 0, 3);
#pragma unroll
      for (int j = 0; j < NSUB; ++j) {
        v2f bf;
        bf[0] = Bl[j * 16];
        bf[1] = Bl[LDB + j * 16];
        acc[j] = __builtin_amdgcn_wmma_f32_16x16x4_f32(
            false, a, false, bf, (short)0, acc[j], false, false);
      }
      Bl += (long)4 * LDB;
    }
  } else {
    long roff = (long)(2 * kh) * LDB;            // element offset of lane's B row
    for (int k0 = 0; k0 < Kmain; k0 += 4) {
      v2f a;
      a[0] = Alds[(k0 + 2 * kh) * 16 + m16];
      a[1] = Alds[(k0 + 2 * kh + 1) * 16 + m16];
      if (k0 + 19 < K)
        __builtin_prefetch(&Bb[roff + (long)16 * LDB + ncl[0]], 0, 3);
#pragma unroll
      for (int j = 0; j < NSUB; ++j) {
        v2f bf;
        bf[0] = Bb[roff + ncl[j]];
        bf[1] = Bb[roff + LDB + ncl[j]];
        acc[j] = __builtin_amdgcn_wmma_f32_16x16x4_f32(
            false, a, false, bf, (short)0, acc[j], false, false);
      }
      roff += (long)4 * LDB;
    }
  }

  // -------- tail (only if K % 4 != 0): clamped loads, zero-padded A --------
  if constexpr ((K & 3) != 0) {
    int ka = Kmain + 2 * kh;
    v2f a;
    a[0] = Alds[ka * 16 + m16];                  // zero for k >= K
    a[1] = Alds[(ka + 1) * 16 + m16];
    int kb0 = (ka     < K) ? ka       : (K - 1);
    int kb1 = (ka + 1 < K) ? (ka + 1) : (K - 1);
#pragma unroll
    for (int j = 0; j < NSUB; ++j) {
      v2f bf;
      bf[0] = Bb[(long)kb0 * LDB + ncl[j]];
      bf[1] = Bb[(long)kb1 * LDB + ncl[j]];
      acc[j] = __builtin_amdgcn_wmma_f32_16x16x4_f32(
          false, a, false, bf, (short)0, acc[j], false, false);
    }
  }

  // C/D layout: col = lane&15; VGPR r holds row r (lanes 0-15) / r+8 (16-31).
#pragma unroll
  for (int j = 0; j < NSUB; ++j) {
    int col = nbase + 16 * j + m16;
    if (NDIV || col < N) {
#pragma unroll
      for (int r = 0; r < 8; ++r) {
        int row = mt * 16 + kh * 8 + r;
        if (row < M) Cb[(long)row * LDC + col] = acc[j][r];
      }
    }
  }
}

// ---------------------------------------------------------------------------
// Kernel 3: per-(b,o) softmax over 273 channels, in place.  Adds the -1e9
// invalid-channel offset on the fly (exp(-1e9 - max) == 0 in fp32, matching
// the reference's NEG_INF semantics).
// ---------------------------------------------------------------------------
__global__ __launch_bounds__(512)
void softmax_kernel(float* __restrict__ scores, const float* __restrict__ positions) {
  __shared__ float red[512];
  int o = blockIdx.x, b = blockIdx.y;
  int tid = threadIdx.x;
  float* row = scores + ((long)b * CHOUT + o) * CC;

  float v = -3.0e38f;
  if (tid < CC) {
    float s = row[tid];
    float px = positions[((long)b * CC + tid) * 2 + 0];
    float py = positions[((long)b * CC + tid) * 2 + 1];
    if (px == -0.1f && py == -0.1f) s += -1.0e9f;
    v = s;
  }
  red[tid] = v; __syncthreads();
  for (int s2 = 256; s2 > 0; s2 >>= 1) {
    if (tid < s2) red[tid] = fmaxf(red[tid], red[tid + s2]);
    __syncthreads();
  }
  float mx = red[0]; __syncthreads();

  float e = (tid < CC) ? expf(v - mx) : 0.0f;
  red[tid] = e; __syncthreads();
  for (int s2 = 256; s2 > 0; s2 >>= 1) {
    if (tid < s2) red[tid] += red[tid + s2];
    __syncthreads();
  }
  float inv = 1.0f / red[0];
  if (tid < CC) row[tid] = e * inv;
}

// ---------------------------------------------------------------------------
// Launch: emb -> scores GEMM (WMMA) -> softmax -> output GEMM (WMMA)
// Workspace: [emb_t: 32*242*273 f32 ~8.5MB][scores/weights: 32*270*273 ~9.4MB]
// ---------------------------------------------------------------------------
extern "C" void kernel_launch(void* const* d_in, const int* in_sizes, int n_in,
                              void* d_out, int out_size, void* d_ws, size_t ws_size,
                              hipStream_t stream) {
  (void)in_sizes; (void)n_in; (void)out_size; (void)ws_size;
  const float* meg           = (const float*)d_in[0];   // [32,273,4096]
  const float* positions     = (const float*)d_in[1];   // [32,273,2]
  const int*   subject_index = (const int*)d_in[2];     // [32]
  const float* heads         = (const float*)d_in[3];   // [200,270,242]
  float* out   = (float*)d_out;                         // [32,270,4096]
  float* emb_t = (float*)d_ws;                          // [32,242,273]
  float* scores = emb_t + (size_t)BB * DD * CC;         // [32,270,273]

  // 1) Fourier embedding (transposed layout).
  emb_kernel<<<dim3(BB * DD), 288, 0, stream>>>(positions, emb_t);

  // 2) scores[b] = heads[subject[b]] (270x242) @ emb_t[b] (242x273)
  constexpr int BLK_N = 512;
  wmma_gemm_f32<DD, CC, CC, CHOUT, CC, DD, true>
      <<<dim3((CC + BLK_N - 1) / BLK_N, (CHOUT + 15) / 16, BB), 256, 0, stream>>>(
          heads, subject_index, emb_t, scores);

  // 3) softmax over channels (adds invalid-channel offset), in place.
  softmax_kernel<<<dim3(CHOUT, BB), 512, 0, stream>>>(scores, positions);

  // 4) out[b] = weights[b] (270x273) @ meg[b] (273x4096)   -- dominant GEMM
  wmma_gemm_f32<CC, TT, TT, CHOUT, TT, CC, false>
      <<<dim3(TT / BLK_N, (CHOUT + 15) / 16, BB), 256, 0, stream>>>(
          scores, nullptr, meg, out);
}